// GrayscaleErosion2D_51797305589786
// MI455X (gfx1250) — compile-verified
//
#include <hip/hip_runtime.h>

// Grayscale 5x5 morphological erosion, image (32,3,1024,1024) f32.
// Bandwidth-bound stencil: LDS-tiled, async global->LDS fill (CDNA5 path),
// register-blocked 4x8 outputs per thread, min-plus accumulation.
// Roofline: ~805 MB HBM traffic -> ~35 us at 23.3 TB/s; compute (~50 VALU
// ops/output) is <1 us, so everything is organized around the memory path.

#define TX 64            // tile width  (outputs)
#define TY 128           // tile height (outputs)
#define KR 5             // kernel size
#define PD 2             // pad
#define LDSW (TX + 2*PD) // 68 floats per LDS row (x4B = 272B, 16B-aligned rows)
#define LDSH (TY + 2*PD) // 132 rows
#define NTHREADS 256     // 8 waves (wave32)

#if __has_builtin(__builtin_amdgcn_global_load_async_to_lds_b32)
#define HAVE_ASYNC_LDS 1
#else
#define HAVE_ASYNC_LDS 0
#endif

typedef __attribute__((address_space(1))) int GInt;      // global int
typedef __attribute__((address_space(3))) int LInt;      // LDS int
typedef float __attribute__((ext_vector_type(4))) v4f;   // native float4

__global__ __launch_bounds__(NTHREADS)
void erosion5x5_gfx1250(const float* __restrict__ img,
                        const float* __restrict__ filt,
                        float* __restrict__ out,
                        int H, int W) {
  __shared__ float tile[LDSH * LDSW];

  const int tid   = threadIdx.x;
  const int bx    = blockIdx.x;
  const int by    = blockIdx.y;
  const int plane = blockIdx.z;

  const int X0  = bx * TX;
  const int Y0  = by * TY;
  const int gx0 = X0 - PD;
  const int gy0 = Y0 - PD;

  const float* __restrict__ ip = img + (size_t)plane * H * W;
  float*       __restrict__ op = out + (size_t)plane * H * W;

  const bool interior = (gx0 >= 0) && (gy0 >= 0) &&
                        (gx0 + LDSW <= W) && (gy0 + LDSH <= H);

  if (interior) {
#if HAVE_ASYNC_LDS
    // CDNA5 async global -> LDS copy (ASYNCcnt-tracked, no VGPR round trip).
    for (int e = tid; e < LDSH * LDSW; e += NTHREADS) {
      const int r = e / LDSW;
      const int c = e - r * LDSW;
      float* gp = const_cast<float*>(ip) + (size_t)(gy0 + r) * W + (gx0 + c);
      __builtin_amdgcn_global_load_async_to_lds_b32(
          (GInt*)gp, (LInt*)&tile[e], /*offset=*/0, /*cpol=*/0);
    }
#if __has_builtin(__builtin_amdgcn_s_wait_asynccnt)
    __builtin_amdgcn_s_wait_asynccnt(0);
#else
    asm volatile("s_wait_asynccnt 0" ::: "memory");
#endif
#else
    for (int e = tid; e < LDSH * LDSW; e += NTHREADS) {
      const int r = e / LDSW;
      const int c = e - r * LDSW;
      tile[e] = ip[(size_t)(gy0 + r) * W + (gx0 + c)];
    }
#endif
  } else {
    // Border tile: guarded fill, +inf is the identity for min.
    for (int e = tid; e < LDSH * LDSW; e += NTHREADS) {
      const int r  = e / LDSW;
      const int c  = e - r * LDSW;
      const int gy = gy0 + r;
      const int gx = gx0 + c;
      float v = __builtin_inff();
      if (gy >= 0 && gy < H && gx >= 0 && gx < W)
        v = ip[(size_t)gy * W + gx];
      tile[e] = v;
    }
  }
  __syncthreads();

  // Filter is wave-uniform -> scalar loads, lives in SGPRs.
  float fw[KR][KR];
#pragma unroll
  for (int i = 0; i < KR; ++i)
#pragma unroll
    for (int j = 0; j < KR; ++j)
      fw[i][j] = filt[i * KR + j];

  // Thread computes a 4-wide x 8-tall output block.
  const int cx = (tid & 15) * 4;  // 16 threads across x -> 64 cols
  const int cy = (tid >> 4) * 8;  // 16 threads down  y -> 128 rows

  float acc[8][4];
#pragma unroll
  for (int oy = 0; oy < 8; ++oy)
#pragma unroll
    for (int ox = 0; ox < 4; ++ox)
      acc[oy][ox] = __builtin_inff();

  // Slide over the 12 input rows this block needs; each row is two aligned
  // ds_load_b128 (8 floats) and feeds up to 5 output rows.
#pragma unroll
  for (int rr = 0; rr < 12; ++rr) {
    const v4f* lp =
        reinterpret_cast<const v4f*>(&tile[(cy + rr) * LDSW + cx]);
    const v4f A = lp[0];
    const v4f B = lp[1];
    float v[8] = {A.x, A.y, A.z, A.w, B.x, B.y, B.z, B.w};
#pragma unroll
    for (int oy = 0; oy < 8; ++oy) {
      const int i = rr - oy;          // filter row for this (input,output) pair
      if (i < 0 || i >= KR) continue; // folds away: rr is compile-time
#pragma unroll
      for (int ox = 0; ox < 4; ++ox)
#pragma unroll
        for (int j = 0; j < KR; ++j)
          acc[oy][ox] = __builtin_fminf(acc[oy][ox], v[ox + j] - fw[i][j]);
    }
  }

  // Streaming output: written once, never re-read -> non-temporal (TH=NT)
  // 128-bit stores keep L2 free for the input halo reuse between tiles.
#pragma unroll
  for (int oy = 0; oy < 8; ++oy) {
    v4f st = {acc[oy][0], acc[oy][1], acc[oy][2], acc[oy][3]};
    v4f* dst =
        reinterpret_cast<v4f*>(&op[(size_t)(Y0 + cy + oy) * W + (X0 + cx)]);
    __builtin_nontemporal_store(st, dst);
  }
}

extern "C" void kernel_launch(void* const* d_in, const int* in_sizes, int n_in,
                              void* d_out, int out_size, void* d_ws, size_t ws_size,
                              hipStream_t stream) {
  const float* img  = (const float*)d_in[0];
  const float* filt = (const float*)d_in[1];
  float*       out  = (float*)d_out;

  const int H = 1024, W = 1024;
  const int planes = in_sizes[0] / (H * W);  // 32*3 = 96

  dim3 grid(W / TX, H / TY, planes);         // (16, 8, 96)
  erosion5x5_gfx1250<<<grid, NTHREADS, 0, stream>>>(img, filt, out, H, W);
}